// SimpChamferLoss_54992761258145
// MI455X (gfx1250) — compile-verified
//
#include <hip/hip_runtime.h>
#include <hip/hip_bf16.h>

typedef __attribute__((ext_vector_type(2))) float v2f;
typedef __attribute__((ext_vector_type(8))) float v8f;

#define ROWS_PER_BLOCK 128           // 8 waves * 16 query rows
#define COLS_PER_CHUNK 1024          // 64 tiles of 16 reference cols per wave
#define TILES_PER_WAVE (COLS_PER_CHUNK / 16)

__device__ __forceinline__ unsigned f2u(float f) { return __float_as_uint(f); }
__device__ __forceinline__ float u2f(unsigned u) { return __uint_as_float(u); }

// Raw v_min_num / v_min3_num without LLVM's input-canonicalize
// (all values here are finite, so NaN semantics are irrelevant).
__device__ __forceinline__ float fmin_raw(float a, float b) {
    float r;
    asm("v_min_num_f32 %0, %1, %2" : "=v"(r) : "v"(a), "v"(b));
    return r;
}
__device__ __forceinline__ float fmin3_raw(float a, float b, float c) {
    float r;
    asm("v_min3_num_f32 %0, %1, %2, %3" : "=v"(r) : "v"(a), "v"(b), "v"(c));
    return r;
}

// ---------------------------------------------------------------------------
// Kernel 0: init min arrays and precompute lane-half-swizzled reference
// operands:  lo[j] = (x, y),  hi[j] = (z, |r|^2).
// With this layout a wave's B matrix for one 16x16x4 WMMA tile is a single
// coalesced b64 load per lane (lanes 0-15 read lo[], lanes 16-31 read hi[]).
// ---------------------------------------------------------------------------
__global__ void chamfer_prep_kernel(const float* __restrict__ P,
                                    const float* __restrict__ G,
                                    float2* __restrict__ Plo, float2* __restrict__ Phi,
                                    float2* __restrict__ Glo, float2* __restrict__ Ghi,
                                    unsigned* __restrict__ minrow,
                                    unsigned* __restrict__ mincol,
                                    int M, int N) {
    int i = blockIdx.x * blockDim.x + threadIdx.x;
    if (i < M) {
        float x = P[i], y = P[M + i], z = P[2 * M + i];
        Plo[i] = make_float2(x, y);
        Phi[i] = make_float2(z, x * x + y * y + z * z);
        minrow[i] = 0x7F800000u;
    }
    if (i < N) {
        float x = G[i], y = G[N + i], z = G[2 * N + i];
        Glo[i] = make_float2(x, y);
        Ghi[i] = make_float2(z, x * x + y * y + z * z);
        mincol[i] = 0x7F800000u;
    }
}

// ---------------------------------------------------------------------------
// Kernel 1 (run twice, roles swapped): NN distances query set Q -> ref set R
// via V_WMMA_F32_16X16X4_F32 with norms folded into K=4:
//   A(16x4) row i = (-2qx, -2qy, -2qz, 1)
//   B(4x16) col j = ( rx,  ry,  rz, |r|^2)   (precomputed, packed)
//   C seeded with |q|^2 per row  =>  D = |q|^2 - 2 q.r + |r|^2 = d2 tile.
// Tiles processed in pairs: per pair 2 b64 loads + 2 WMMA + 8 v_min3
// (running min folded with both tiles in one VOP3).
// Clamp to 0 is hoisted out of the loop (commutes with min).
// ---------------------------------------------------------------------------
__global__ void __launch_bounds__(256)
chamfer_nn_kernel(const float2* __restrict__ Qlo, const float2* __restrict__ Qhi,
                  const float2* __restrict__ Rlo, const float2* __restrict__ Rhi,
                  unsigned* __restrict__ minQ, int nq, int nr) {
    const int tid    = threadIdx.x;
    const int waveId = tid >> 5;
    const int lane   = tid & 31;
    const int lane16 = lane & 15;
    const bool hi    = lane >= 16;

    const int rowBase = blockIdx.y * ROWS_PER_BLOCK + waveId * 16;
    const int colBase = blockIdx.x * COLS_PER_CHUNK;

    // --- this wave's 16 query points (A operand + C seed) ---
    const int row = rowBase + lane16;
    const float2 ql = Qlo[row];
    const float2 qh = Qhi[row];
    const float qn2 = qh.y;

    // A: lanes 0-15 hold K={0,1}, lanes 16-31 hold K={2,3}, M = lane%16
    v2f a;
    a.x = hi ? (-2.0f * qh.x) : (-2.0f * ql.x);
    a.y = hi ? 1.0f           : (-2.0f * ql.y);

    // C seed: VGPR v holds row v (+8 for hi lanes); |q|^2 lives in lane (row%16)
    const int srcBase = hi ? 8 : 0;
    v8f cinit;
#pragma unroll
    for (int v = 0; v < 8; ++v)
        cinit[v] = __shfl(qn2, srcBase + v, 32);

    float rmin[8];
#pragma unroll
    for (int v = 0; v < 8; ++v) rmin[v] = __builtin_inff();

    // B operand source: one packed b64 per lane per tile (pointer hoisted)
    const float2* __restrict__ Bp = (hi ? Rhi : Rlo) + colBase + lane16;

#pragma unroll 2
    for (int t = 0; t < TILES_PER_WAVE; t += 2) {
        const float2 bb0 = Bp[(t + 0) * 16];
        const float2 bb1 = Bp[(t + 1) * 16];
        v2f b0, b1;
        b0.x = bb0.x; b0.y = bb0.y;
        b1.x = bb1.x; b1.y = bb1.y;

        v8f d0 = __builtin_amdgcn_wmma_f32_16x16x4_f32(
            false, a, false, b0, (short)0, cinit, false, false);
        v8f d1 = __builtin_amdgcn_wmma_f32_16x16x4_f32(
            false, a, false, b1, (short)0, cinit, false, false);

        // fold both tiles into the running row min in one VOP3 per element
#pragma unroll
        for (int v = 0; v < 8; ++v)
            rmin[v] = fmin3_raw(rmin[v], d0[v], d1[v]);
    }

    // reduce row mins across the 16 lanes of each half (one-time cost)
#pragma unroll
    for (int off = 8; off >= 1; off >>= 1) {
#pragma unroll
        for (int v = 0; v < 8; ++v)
            rmin[v] = fmin_raw(rmin[v], __shfl_xor(rmin[v], off, 32));
    }
    if (lane16 == 0) {
#pragma unroll
        for (int v = 0; v < 8; ++v) {
            float m = fmaxf(rmin[v], 0.0f);   // hoisted clamp (nonneg for uint-min)
            atomicMin(&minQ[rowBase + srcBase + v], f2u(m));
        }
    }
}

// ---------------------------------------------------------------------------
// Kernel 2: single-block final reduction -> [cd, fscore(th0), fscore(th1)]
// ---------------------------------------------------------------------------
__global__ void __launch_bounds__(512)
chamfer_final_kernel(const unsigned* __restrict__ minrow,
                     const unsigned* __restrict__ mincol,
                     const float* __restrict__ threshes,
                     float* __restrict__ out, int M, int N) {
    __shared__ float sSumF[512], sSumB[512];
    __shared__ float sC0F[512], sC1F[512], sC0B[512], sC1B[512];

    const int tid = threadIdx.x;
    const float t0 = threshes[0];
    const float t1 = threshes[1];

    float sumF = 0.0f, sumB = 0.0f;
    float c0f = 0.0f, c1f = 0.0f, c0b = 0.0f, c1b = 0.0f;

    for (int i = tid; i < M; i += 512) {
        float fd = sqrtf(u2f(minrow[i]));     // stored values already >= 0
        sumF += fd;
        c0f += (fd <= t0) ? 1.0f : 0.0f;
        c1f += (fd <= t1) ? 1.0f : 0.0f;
    }
    for (int i = tid; i < N; i += 512) {
        float bd = sqrtf(u2f(mincol[i]));
        sumB += bd;
        c0b += (bd <= t0) ? 1.0f : 0.0f;
        c1b += (bd <= t1) ? 1.0f : 0.0f;
    }

    sSumF[tid] = sumF; sSumB[tid] = sumB;
    sC0F[tid] = c0f;   sC1F[tid] = c1f;
    sC0B[tid] = c0b;   sC1B[tid] = c1b;
    __syncthreads();

    for (int s = 256; s > 0; s >>= 1) {
        if (tid < s) {
            sSumF[tid] += sSumF[tid + s];
            sSumB[tid] += sSumB[tid + s];
            sC0F[tid]  += sC0F[tid + s];
            sC1F[tid]  += sC1F[tid + s];
            sC0B[tid]  += sC0B[tid + s];
            sC1B[tid]  += sC1B[tid + s];
        }
        __syncthreads();
    }

    if (tid == 0) {
        const float Mf = (float)M;
        out[0] = 0.5f * (sSumF[0] / Mf) + 0.5f * (sSumB[0] / Mf);
        float cf[2] = { sC0F[0], sC1F[0] };
        float cb[2] = { sC0B[0], sC1B[0] };
#pragma unroll
        for (int t = 0; t < 2; ++t) {
            float p = (100.0f / Mf) * cf[t];
            float r = (100.0f / Mf) * cb[t];
            out[1 + t] = 2.0f * p * r / (p + r + 1e-8f);
        }
    }
}

// ---------------------------------------------------------------------------
extern "C" void kernel_launch(void* const* d_in, const int* in_sizes, int n_in,
                              void* d_out, int out_size, void* d_ws, size_t ws_size,
                              hipStream_t stream) {
    const float* P  = (const float*)d_in[0];   // [1,3,M]
    const float* G  = (const float*)d_in[1];   // [1,3,N]
    const float* th = (const float*)d_in[2];   // [2]
    float* out = (float*)d_out;

    const int M = in_sizes[0] / 3;
    const int N = in_sizes[1] / 3;

    // workspace layout (8B-aligned chunks):
    char* ws = (char*)d_ws;
    float2* Plo = (float2*)ws;               ws += (size_t)M * sizeof(float2);
    float2* Phi = (float2*)ws;               ws += (size_t)M * sizeof(float2);
    float2* Glo = (float2*)ws;               ws += (size_t)N * sizeof(float2);
    float2* Ghi = (float2*)ws;               ws += (size_t)N * sizeof(float2);
    unsigned* minrow = (unsigned*)ws;        ws += (size_t)M * sizeof(unsigned);
    unsigned* mincol = (unsigned*)ws;

    // 0) precompute packed operands + init min buffers (ws is poisoned)
    {
        int total = (M > N ? M : N);
        int blocks = (total + 255) / 256;
        chamfer_prep_kernel<<<blocks, 256, 0, stream>>>(
            P, G, Plo, Phi, Glo, Ghi, minrow, mincol, M, N);
    }

    // 1a) forward pass: P queries vs G references -> minrow
    {
        dim3 grid(N / COLS_PER_CHUNK, M / ROWS_PER_BLOCK);
        chamfer_nn_kernel<<<grid, 256, 0, stream>>>(Plo, Phi, Glo, Ghi, minrow, M, N);
    }
    // 1b) backward pass: G queries vs P references -> mincol
    {
        dim3 grid(M / COLS_PER_CHUNK, N / ROWS_PER_BLOCK);
        chamfer_nn_kernel<<<grid, 256, 0, stream>>>(Glo, Ghi, Plo, Phi, mincol, N, M);
    }

    // 2) final scalar metrics
    chamfer_final_kernel<<<1, 512, 0, stream>>>(minrow, mincol, th, out, M, N);
}